// GraphAttentionLayer_68719476736535
// MI455X (gfx1250) — compile-verified
//
#include <hip/hip_runtime.h>
#include <hip/hip_bf16.h>

// GAT layer for MI455X (gfx1250, wave32).
// Bandwidth-bound: 134MB attention write + 134MB adj read dominate (~11.5us
// floor at 23.3 TB/s). adj is compressed to a 4MB bitmask so it is read from
// HBM exactly once. Matmuls use V_WMMA_F32_16X16X4_F32 (fp32-exact vs the
// reference; at 4.3 GFLOP total the matmul rate is irrelevant). Wh tiles are
// staged into LDS with GLOBAL_LOAD_ASYNC_TO_LDS_B128 (ASYNCcnt) overlapped
// with the attention-tile exp/store work.

#define B_DIM 8
#define N_DIM 2048
#define F_IN 128
#define F_OUT 64
#define ALPHA 0.2f
#define WPB 2   // waves per block in k_att (512 blocks in flight)

typedef __attribute__((ext_vector_type(2))) float v2f;
typedef __attribute__((ext_vector_type(8))) float v8f;
typedef int vsi4 __attribute__((vector_size(16)));  // matches builtin proto

#if __has_builtin(__builtin_amdgcn_global_load_async_to_lds_b128)
#define HAVE_ASYNC_LDS 1
#else
#define HAVE_ASYNC_LDS 0
#endif

__device__ __forceinline__ unsigned ballot32(bool p) {
#if __has_builtin(__builtin_amdgcn_ballot_w32)
  return __builtin_amdgcn_ballot_w32(p);
#else
  return (unsigned)__ballot(p);
#endif
}

#if HAVE_ASYNC_LDS
__device__ __forceinline__ void wait_async0() {
#if __has_builtin(__builtin_amdgcn_s_wait_asynccnt)
  __builtin_amdgcn_s_wait_asynccnt(0);
#else
  asm volatile("s_wait_asynccnt 0x0" ::: "memory");
#endif
}
#endif

// ---------------------------------------------------------------------------
// Kernel 1: Wh = h @ W  (WMMA f32 16x16x4), fused f1 = Wh@a1, f2 = Wh@a2.
// 4 waves/block, one 16-row tile per wave. W lives in LDS (padded stride 65).
// ---------------------------------------------------------------------------
__global__ __launch_bounds__(128) void k_wh(
    const float* __restrict__ h, const float* __restrict__ W,
    const float* __restrict__ a1, const float* __restrict__ a2,
    float* __restrict__ Wh, float* __restrict__ f1, float* __restrict__ f2) {
  __shared__ float Ws[F_IN][F_OUT + 1];  // stride 65: no bank conflicts
  const int t = threadIdx.x;
  const int lane = t & 31;
  const int w = t >> 5;
  for (int q = t; q < F_IN * F_OUT; q += 128) Ws[q >> 6][q & 63] = W[q];
  __syncthreads();

  const int row0 = (blockIdx.x * 4 + w) * 16;  // over B*N rows
  const int mrow = lane & 15;                  // M index (A/B/C lane mapping)
  const int khalf = (lane >> 4) << 1;          // K pair selector for A/B regs

  v8f acc[4] = {};  // 4 col-tiles of 16 -> 64 output cols
#pragma unroll 4
  for (int k0 = 0; k0 < F_IN; k0 += 4) {
    // A: 16x4 f32 tile of h (ISA layout: lane<16 K=0,1 ; lane>=16 K=2,3)
    v2f a = *(const v2f*)(h + (size_t)(row0 + mrow) * F_IN + k0 + khalf);
#pragma unroll
    for (int c = 0; c < 4; ++c) {
      v2f bv;
      bv.x = Ws[k0 + khalf][c * 16 + mrow];
      bv.y = Ws[k0 + khalf + 1][c * 16 + mrow];
      acc[c] = __builtin_amdgcn_wmma_f32_16x16x4_f32(
          false, a, false, bv, (short)0, acc[c], false, false);
    }
  }

  // Store Wh and reduce f1/f2. C layout: reg r -> row r (lanes 0-15) / r+8.
  float a1v[4], a2v[4];
#pragma unroll
  for (int c = 0; c < 4; ++c) {
    a1v[c] = a1[c * 16 + mrow];
    a2v[c] = a2[c * 16 + mrow];
  }
#pragma unroll
  for (int r = 0; r < 8; ++r) {
    const int rowG = row0 + r + ((lane >> 4) << 3);
    float s1 = 0.f, s2 = 0.f;
#pragma unroll
    for (int c = 0; c < 4; ++c) {
      float v = acc[c][r];
      Wh[(size_t)rowG * F_OUT + c * 16 + mrow] = v;
      s1 += v * a1v[c];
      s2 += v * a2v[c];
    }
#pragma unroll
    for (int off = 8; off; off >>= 1) {  // reduce within lane%16 group
      s1 += __shfl_xor(s1, off);
      s2 += __shfl_xor(s2, off);
    }
    if (mrow == 0) {
      f1[rowG] = s1;
      f2[rowG] = s2;
    }
  }
}

// ---------------------------------------------------------------------------
// Kernel 2: per-row online softmax stats (max m, sum l) over masked
// LeakyReLU(f1[i]+f2[j]); packs adj into a bitmask (adj read once from HBM).
// One wave per row, 8 waves/block; f2 row cached in LDS.
// ---------------------------------------------------------------------------
__global__ __launch_bounds__(256) void k_stats(
    const int* __restrict__ adj, const float* __restrict__ f1,
    const float* __restrict__ f2, float* __restrict__ mrow,
    float* __restrict__ linv, unsigned* __restrict__ adjbits) {
  __shared__ float f2s[N_DIM];
  const int b = blockIdx.y;
  for (int q = threadIdx.x; q < N_DIM; q += 256) f2s[q] = f2[b * N_DIM + q];
  __syncthreads();

  const int lane = threadIdx.x & 31;
  const int row = blockIdx.x * 8 + (threadIdx.x >> 5);
  const size_t rbase = (size_t)b * N_DIM + row;
  const int* arow = adj + rbase * N_DIM;
  unsigned* brow = adjbits + rbase * (N_DIM / 32);
  const float f1r = f1[rbase];

  float m = -3.0e38f, l = 0.f;
  unsigned wbuf = 0;
  for (int k = 0; k < N_DIM / 32; ++k) {
    const int j = k * 32 + lane;
    const int av = arow[j];
    float e = f1r + f2s[j];
    e = e >= 0.f ? e : ALPHA * e;
    const unsigned word = ballot32(av != 0);
    if ((k & 31) == lane) wbuf = word;          // lane (k%32) keeps word k
    if ((k & 31) == 31) brow[(k - 31) + lane] = wbuf;  // coalesced flush
    const float em = av ? e : -3.0e38f;
    const float mn = fmaxf(m, em);
    l = l * __expf(m - mn) + (av ? __expf(em - mn) : 0.f);
    m = mn;
  }
  // wave-level (m,l) merge
#pragma unroll
  for (int off = 16; off; off >>= 1) {
    const float mo = __shfl_xor(m, off);
    const float lo = __shfl_xor(l, off);
    const float mn = fmaxf(m, mo);
    l = l * __expf(m - mn) + lo * __expf(mo - mn);
    m = mn;
  }
  if (lane == 0) {
    mrow[rbase] = m;
    linv[rbase] = 1.0f / l;
  }
}

// ---------------------------------------------------------------------------
// Kernel 3: attention = exp(e - m)/l (written to d_out, coalesced) and
// h_prime = attention @ Wh via WMMA f32 16x16x4.
// One wave per 16-row i-tile; WPB waves share a 32x64 Wh LDS tile per j-step.
// Wh tiles staged with async global->LDS DMA, overlapped with att compute.
// ---------------------------------------------------------------------------
__global__ __launch_bounds__(32 * WPB) void k_att(
    const float* __restrict__ Wh, const float* __restrict__ f1,
    const float* __restrict__ f2, const float* __restrict__ mrow,
    const float* __restrict__ linv, const unsigned* __restrict__ adjbits,
    float* __restrict__ hprime, float* __restrict__ attOut) {
  __shared__ float f2s[N_DIM];              // 8KB
  __shared__ float WhS[32][F_OUT + 4];      // stride 68
  __shared__ float attS[WPB][16][34];       // stride 34 (even -> b64-aligned)
  const int t = threadIdx.x, lane = t & 31, w = t >> 5;
  const int b = blockIdx.y;
  const int i0 = (blockIdx.x * WPB + w) * 16;

  for (int q = t; q < N_DIM; q += 32 * WPB) f2s[q] = f2[b * N_DIM + q];

  // per-row params: lane (lane&15) owns row i0+(lane&15); broadcast via shfl
  const int rloc = lane & 15;
  const size_t prow = (size_t)b * N_DIM + i0 + rloc;
  const float mv = mrow[prow];
  const float iv = linv[prow];
  const float f1v = f1[prow];
  const unsigned* bitsBase = adjbits + prow * (N_DIM / 32);
  __syncthreads();

  const int mcol = lane & 15;
  const int khalf = (lane >> 4) << 1;
  v8f acc[4] = {};

  for (int jt = 0; jt < N_DIM / 32; ++jt) {
    const int j0 = jt * 32;
    __syncthreads();  // previous-iteration WhS readers are done
    const float4* src = (const float4*)(Wh + ((size_t)b * N_DIM + j0) * F_OUT);
#if HAVE_ASYNC_LDS
    // Async DMA of the 32x64 Wh tile straight into LDS (no VGPR round-trip);
    // overlaps with the attention-tile exp/mask/store work below.
#pragma unroll
    for (int q = t; q < 512; q += 32 * WPB) {
      const int rr = q >> 4, cc = (q & 15) << 2;
      __builtin_amdgcn_global_load_async_to_lds_b128(
          (__attribute__((address_space(1))) vsi4*)(src + q),
          (__attribute__((address_space(3))) vsi4*)&WhS[rr][cc],
          0, 0);
    }
#else
#pragma unroll
    for (int q = t; q < 512; q += 32 * WPB) {
      float4 v = src[q];
      const int rr = q >> 4, cc = (q & 15) << 2;
      WhS[rr][cc] = v.x;
      WhS[rr][cc + 1] = v.y;
      WhS[rr][cc + 2] = v.z;
      WhS[rr][cc + 3] = v.w;
    }
#endif
    if (jt + 1 < N_DIM / 32)  // global_prefetch next tile
      __builtin_prefetch(Wh + ((size_t)b * N_DIM + j0 + 32) * F_OUT + t * 4, 0, 1);

    // recompute attention tile [16 rows][32 j] from bitmask + scalars
    const unsigned wword = bitsBase[jt];
    const float f2j = f2s[j0 + lane];
#pragma unroll
    for (int r = 0; r < 16; ++r) {
      const unsigned word_r = __shfl(wword, r);
      const float m_r = __shfl(mv, r);
      const float il_r = __shfl(iv, r);
      const float f1_r = __shfl(f1v, r);
      float e = f1_r + f2j;
      e = e >= 0.f ? e : ALPHA * e;
      const float att = ((word_r >> lane) & 1u) ? __expf(e - m_r) * il_r : 0.f;
      attOut[((size_t)b * N_DIM + i0 + r) * N_DIM + j0 + lane] = att;  // coalesced
      attS[w][r][lane] = att;
    }
#if HAVE_ASYNC_LDS
    wait_async0();   // this wave's DMA done; barrier covers the other waves
#endif
    __syncthreads();  // WhS ready + attS visible across lanes

    // h_prime accumulation: A = att (16xK), B = Wh (Kx64), K in chunks of 4
#pragma unroll
    for (int k0 = 0; k0 < 32; k0 += 4) {
      v2f a;
      a.x = attS[w][mcol][k0 + khalf];
      a.y = attS[w][mcol][k0 + khalf + 1];
#pragma unroll
      for (int c = 0; c < 4; ++c) {
        v2f bv;
        bv.x = WhS[k0 + khalf][c * 16 + mcol];
        bv.y = WhS[k0 + khalf + 1][c * 16 + mcol];
        acc[c] = __builtin_amdgcn_wmma_f32_16x16x4_f32(
            false, a, false, bv, (short)0, acc[c], false, false);
      }
    }
  }

  // store h_prime (C layout: reg r -> row r / r+8, lane%16 -> col within tile)
#pragma unroll
  for (int c = 0; c < 4; ++c)
#pragma unroll
    for (int r = 0; r < 8; ++r) {
      const int rowG = i0 + r + ((lane >> 4) << 3);
      hprime[((size_t)b * N_DIM + rowG) * F_OUT + c * 16 + mcol] = acc[c][r];
    }
}

extern "C" void kernel_launch(void* const* d_in, const int* in_sizes, int n_in,
                              void* d_out, int out_size, void* d_ws, size_t ws_size,
                              hipStream_t stream) {
  (void)in_sizes; (void)n_in; (void)out_size; (void)ws_size;
  const float* h = (const float*)d_in[0];
  const int* adj = (const int*)d_in[1];
  const float* W = (const float*)d_in[2];
  const float* a1 = (const float*)d_in[3];
  const float* a2 = (const float*)d_in[4];

  float* hprime = (float*)d_out;                                   // [B,N,64]
  float* attOut = hprime + (size_t)B_DIM * N_DIM * F_OUT;          // [B,N,N]

  // workspace layout
  char* ws = (char*)d_ws;
  float* Wh = (float*)ws;                                          // 4 MB
  float* f1 = (float*)(ws + (size_t)B_DIM * N_DIM * F_OUT * 4);
  float* f2 = f1 + (size_t)B_DIM * N_DIM;
  float* mr = f2 + (size_t)B_DIM * N_DIM;
  float* li = mr + (size_t)B_DIM * N_DIM;
  unsigned* adjbits = (unsigned*)(li + (size_t)B_DIM * N_DIM);     // 4 MB

  // 1) Wh, f1, f2   (1024 row-tiles, 4 waves/block)
  k_wh<<<256, 128, 0, stream>>>(h, W, a1, a2, Wh, f1, f2);
  // 2) softmax stats + adj bitmask (one wave per row)
  k_stats<<<dim3(N_DIM / 8, B_DIM), 256, 0, stream>>>(adj, f1, f2, mr, li, adjbits);
  // 3) attention write + h_prime WMMA (512 blocks)
  k_att<<<dim3(N_DIM / 16 / WPB, B_DIM), 32 * WPB, 0, stream>>>(
      Wh, f1, f2, mr, li, adjbits, hprime, attOut);
}